// DeformableConv2d_5265629905356
// MI455X (gfx1250) — compile-verified
//
#include <hip/hip_runtime.h>
#include <hip/hip_bf16.h>

typedef __attribute__((ext_vector_type(2))) float v2f;
typedef __attribute__((ext_vector_type(8))) float v8f;

#define B_  4
#define C_  64
#define H_  128
#define W_  128
#define O_  64
#define KK_ 9
#define KC_ 576              // contraction length = KK_*C_
#define TW  16               // pixel-tile width (N tile)
#define NTILES (B_ * H_ * (W_ / TW))   // 4096

// ---------------- workspace layout (floats) ----------------
// wT   : O_*KC_              = 36864   weight repacked as [o][kk*64+c]
// pgw  : 32*KC_              = 18432   pg_w zero-padded to 32 rows, [co][ci*9+kp]
// DY   : B_*KK_*H_*W_        = 589824
// DX   : same
// MK   : same
#define WS_WT   0
#define WS_PGW  (WS_WT + O_*KC_)
#define WS_DY   (WS_PGW + 32*KC_)
#define WS_DX   (WS_DY + B_*KK_*H_*W_)
#define WS_MK   (WS_DX + B_*KK_*H_*W_)

// ---------------------------------------------------------------------------
// Kernel 0: repack weights for fragment-friendly GEMM A-operands.
// ---------------------------------------------------------------------------
__global__ __launch_bounds__(256) void dcn_prep_kernel(
    const float* __restrict__ weight, const float* __restrict__ pg_w,
    float* __restrict__ wT, float* __restrict__ pgw)
{
    int e = blockIdx.x * 256 + threadIdx.x;
    if (e < O_ * KC_) {
        int o = e / KC_, r = e - o * KC_;
        int kk = r >> 6, c = r & 63;
        wT[e] = weight[o * KC_ + c * KK_ + kk];      // wT[o][kk*64+c]
    }
    if (e < 32 * KC_) {
        int co = e / KC_;
        pgw[e] = (co < 3 * KK_) ? pg_w[e] : 0.0f;    // pad rows 27..31 with 0
    }
}

// ---------------------------------------------------------------------------
// Kernel 1: offset-generating 3x3 conv via WMMA, fused split + sigmoid.
// Block = 64 threads (2 waves), one 16-pixel tile; wave m computes rows
// m*16..m*16+15 of the 32(padded)x16 params tile.
// ---------------------------------------------------------------------------
__global__ __launch_bounds__(64) void dcn_offsets_kernel(
    const float* __restrict__ x, const float* __restrict__ pgw,
    const float* __restrict__ pg_b,
    float* __restrict__ DY, float* __restrict__ DX, float* __restrict__ MK)
{
    __shared__ float smem[KC_ * TW];   // im2col, paired layout; 36 KB

    const int tile = blockIdx.x;
    const int wb = (tile & 7) * TW;
    const int h  = (tile >> 3) & (H_ - 1);
    const int b  = tile >> 10;
    const int t  = threadIdx.x;

    // ---- build im2col: value(K,p), K = ci*9 + kp, kp = (ky+1)*3 + (kx+1)
    for (int i = t; i < KC_ * TW; i += 64) {
        int p = i & 15;
        int K = i >> 4;
        int ci = K / 9, kp = K - ci * 9;
        int yy = h + kp / 3 - 1;
        int xx = wb + p + (kp % 3) - 1;
        float v = 0.0f;
        if (yy >= 0 && yy < H_ && xx >= 0 && xx < W_)
            v = x[((b * C_ + ci) * H_ + yy) * W_ + xx];
        smem[((K >> 1) * TW + p) * 2 + (K & 1)] = v;
    }
    __syncthreads();

    const int wave = t >> 5;
    const int lane = t & 31;
    const int mrow = lane & 15;
    const int kh   = lane >> 4;            // 0: K {0,1}, 1: K {2,3}
    const float* arow = pgw + (wave * 16 + mrow) * KC_ + kh * 2;
    const v2f*  bbase = (const v2f*)smem;

    v8f acc = {};
    #pragma unroll 4
    for (int ks = 0; ks < KC_ / 4; ++ks) {
        v2f a = *(const v2f*)(arow + ks * 4);
        v2f bf = bbase[(ks * 2 + kh) * TW + mrow];
        acc = __builtin_amdgcn_wmma_f32_16x16x4_f32(
                  false, a, false, bf, (short)0, acc, false, false);
    }
    __syncthreads();                       // im2col no longer needed

    // ---- params (+pg_b) -> smem[0..511]
    #pragma unroll
    for (int r = 0; r < 8; ++r) {
        int row = wave * 16 + r + 8 * kh;
        float v = acc[r] + (row < 3 * KK_ ? pg_b[row] : 0.0f);
        smem[row * TW + mrow] = v;
    }
    __syncthreads();

    // ---- dy = params[2k], dx = params[2k+1], mask = sigmoid(params[18+k])
    for (int i = t; i < KK_ * TW; i += 64) {
        int p = i & 15;
        int k = i >> 4;
        float dyv = smem[(2 * k) * TW + p];
        float dxv = smem[(2 * k + 1) * TW + p];
        float mv  = smem[(2 * KK_ + k) * TW + p];
        mv = 1.0f / (1.0f + __expf(-mv));
        int gi = ((b * KK_ + k) * H_ + h) * W_ + wb + p;
        DY[gi] = dyv; DX[gi] = dxv; MK[gi] = mv;
    }
}

// ---------------------------------------------------------------------------
// Kernel 2: bilinear gather (mask folded into corner weights) into LDS,
// then 16x16x576 fp32 WMMA contraction per wave. Block = 128 threads
// (4 waves = 4 M-tiles of output channels), one 16-pixel tile.
// ---------------------------------------------------------------------------
__global__ __launch_bounds__(128) void dcn_main_kernel(
    const float* __restrict__ x, const float* __restrict__ wT,
    const float* __restrict__ bias,
    const float* __restrict__ DY, const float* __restrict__ DX,
    const float* __restrict__ MK, float* __restrict__ out)
{
    __shared__ float smem[KC_ * TW];       // sampled[K=kk*64+c][p], paired

    const int tile = blockIdx.x;
    const int wb = (tile & 7) * TW;
    const int h  = (tile >> 3) & (H_ - 1);
    const int b  = tile >> 10;
    const int t  = threadIdx.x;
    const int HW = H_ * W_;
    const float* xb = x + b * C_ * HW;

    // ---- sampling: 144 (kk,p) coordinate tasks over 128 threads
    for (int idx = t; idx < KK_ * TW; idx += 128) {
        int p  = idx & 15;
        int kk = idx >> 4;
        int w  = wb + p;
        int gi = ((b * KK_ + kk) * H_ + h) * W_ + w;
        float dy = DY[gi], dx = DX[gi], m = MK[gi];
        float ys = (float)(h - 1 + kk / 3) + dy;
        float xs = (float)(w - 1 + kk % 3) + dx;
        float y0f = floorf(ys), x0f = floorf(xs);
        float ly = ys - y0f, lx = xs - x0f;
        int y0 = (int)y0f, x0 = (int)x0f;
        int y1 = y0 + 1,   x1 = x0 + 1;
        float vy0 = (y0 >= 0 && y0 < H_) ? 1.0f : 0.0f;
        float vy1 = (y1 >= 0 && y1 < H_) ? 1.0f : 0.0f;
        float vx0 = (x0 >= 0 && x0 < W_) ? 1.0f : 0.0f;
        float vx1 = (x1 >= 0 && x1 < W_) ? 1.0f : 0.0f;
        int yc0 = min(max(y0, 0), H_ - 1), yc1 = min(max(y1, 0), H_ - 1);
        int xc0 = min(max(x0, 0), W_ - 1), xc1 = min(max(x1, 0), W_ - 1);
        float w00 = (1.0f - ly) * (1.0f - lx) * m * vy0 * vx0;
        float w01 = (1.0f - ly) * lx          * m * vy0 * vx1;
        float w10 = ly * (1.0f - lx)          * m * vy1 * vx0;
        float w11 = ly * lx                   * m * vy1 * vx1;
        const float* p00 = xb + yc0 * W_ + xc0;
        const float* p01 = xb + yc0 * W_ + xc1;
        const float* p10 = xb + yc1 * W_ + xc0;
        const float* p11 = xb + yc1 * W_ + xc1;
        #pragma unroll 4
        for (int c = 0; c < C_; ++c) {
            float v = p00[c * HW] * w00 + p01[c * HW] * w01
                    + p10[c * HW] * w10 + p11[c * HW] * w11;
            int K = (kk << 6) + c;
            smem[((K >> 1) * TW + p) * 2 + (K & 1)] = v;
        }
    }
    __syncthreads();

    // ---- GEMM: wave -> 16 output channels, 144 x v_wmma_f32_16x16x4_f32
    const int wave = t >> 5;
    const int lane = t & 31;
    const int mrow = lane & 15;
    const int kh   = lane >> 4;
    const float* arow = wT + (wave * 16 + mrow) * KC_ + kh * 2;
    const v2f*  bbase = (const v2f*)smem;

    v8f acc = {};
    #pragma unroll 4
    for (int ks = 0; ks < KC_ / 4; ++ks) {
        v2f a  = *(const v2f*)(arow + ks * 4);
        v2f bf = bbase[(ks * 2 + kh) * TW + mrow];
        acc = __builtin_amdgcn_wmma_f32_16x16x4_f32(
                  false, a, false, bf, (short)0, acc, false, false);
    }

    #pragma unroll
    for (int r = 0; r < 8; ++r) {
        int o = wave * 16 + r + 8 * kh;
        out[((b * O_ + o) * H_ + h) * W_ + wb + mrow] = acc[r] + bias[o];
    }
}

// ---------------------------------------------------------------------------
extern "C" void kernel_launch(void* const* d_in, const int* in_sizes, int n_in,
                              void* d_out, int out_size, void* d_ws, size_t ws_size,
                              hipStream_t stream)
{
    const float* x      = (const float*)d_in[0];
    const float* weight = (const float*)d_in[1];
    const float* bias   = (const float*)d_in[2];
    const float* pg_w   = (const float*)d_in[3];
    const float* pg_b   = (const float*)d_in[4];
    float* out = (float*)d_out;

    float* ws  = (float*)d_ws;
    float* wT  = ws + WS_WT;
    float* pgw = ws + WS_PGW;
    float* DY  = ws + WS_DY;
    float* DX  = ws + WS_DX;
    float* MK  = ws + WS_MK;

    dcn_prep_kernel<<<(O_ * KC_ + 255) / 256, 256, 0, stream>>>(weight, pg_w, wT, pgw);
    dcn_offsets_kernel<<<NTILES, 64, 0, stream>>>(x, pgw, pg_b, DY, DX, MK);
    dcn_main_kernel<<<NTILES, 128, 0, stream>>>(x, wT, bias, DY, DX, MK, out);
}